// TriXFFT_53584011985642
// MI455X (gfx1250) — compile-verified
//
#include <hip/hip_runtime.h>

typedef __attribute__((ext_vector_type(2))) float v2f;
typedef __attribute__((ext_vector_type(8))) float v8f;

#define WAVES_PER_BLOCK 2
#define THREADS (WAVES_PER_BLOCK * 32)

// D = A(16x4) * B(4x16) + C, fp32 WMMA (CDNA5 V_WMMA_F32_16X16X4_F32)
__device__ __forceinline__ v8f wmma_k4(v2f a, v2f b, v8f c) {
  return __builtin_amdgcn_wmma_f32_16x16x4_f32(false, a, false, b, (short)0, c,
                                               false, false);
}

__global__ __launch_bounds__(THREADS) void fft1024_wmma(
    const float* __restrict__ xre, const float* __restrict__ xim,
    float* __restrict__ yre, float* __restrict__ yim, int rows) {
  // W1024 twiddle table (cos/sin), built once per block
  __shared__ float twc[1024];
  __shared__ float tws[1024];
  // Stage A -> B staging: [j2][m1][n1] flattened (1024 fp32 per plane per wave)
  __shared__ float sBre[WAVES_PER_BLOCK][1024];
  __shared__ float sBim[WAVES_PER_BLOCK][1024];
  // Stage B -> C staging: [n1][k2] with padded row stride 65 (bank-conflict free)
  __shared__ float sCre[WAVES_PER_BLOCK][16 * 65];
  __shared__ float sCim[WAVES_PER_BLOCK][16 * 65];

  const int tid = threadIdx.x;
  for (int i = tid; i < 1024; i += THREADS) {
    float s, c;
    __sincosf(-6.28318530717958647692f * (float)i * (1.0f / 1024.0f), &s, &c);
    twc[i] = c;
    tws[i] = s;
  }
  __syncthreads();

  const int wave = tid >> 5;
  const int lane = tid & 31;
  int row = blockIdx.x * WAVES_PER_BLOCK + wave;
  if (row >= rows) row = rows - 1;  // duplicate tail work, keeps barriers uniform

  const int m = lane & 15;   // A-operand M row / B column index
  const int hi = lane >> 4;  // lane-half selector

  // F16 DFT matrix as WMMA A-operand chunks: A[m][k] = W16^(m*k),
  // chunk kc covers K = 4kc..4kc+3; lane-half holds K-pairs {0,1}/{2,3}.
  float fre[4][2], fim[4][2], fng[4][2];
#pragma unroll
  for (int kc = 0; kc < 4; ++kc) {
#pragma unroll
    for (int r = 0; r < 2; ++r) {
      const int k = 4 * kc + 2 * hi + r;
      const int t = ((m * k) & 15) << 6;  // 64 * (m*k mod 16) indexes W1024
      fre[kc][r] = twc[t];
      fim[kc][r] = tws[t];
      fng[kc][r] = -tws[t];
    }
  }

  // ---------------- Stage A: radix-4 over m2 + W64 twiddle ----------------
  // n = m2*256 + p, p = m1*16 + n1; this lane owns p = lane*8 .. lane*8+7.
  const float* rr = xre + (size_t)row * 1024;
  const float* ri = xim + (size_t)row * 1024;
  float ar[4][8], ai[4][8];
#pragma unroll
  for (int m2 = 0; m2 < 4; ++m2) {
    const float4* pr = reinterpret_cast<const float4*>(rr + m2 * 256 + lane * 8);
    const float4* pi = reinterpret_cast<const float4*>(ri + m2 * 256 + lane * 8);
    float4 r0 = pr[0], r1 = pr[1];
    float4 i0 = pi[0], i1 = pi[1];
    ar[m2][0] = r0.x; ar[m2][1] = r0.y; ar[m2][2] = r0.z; ar[m2][3] = r0.w;
    ar[m2][4] = r1.x; ar[m2][5] = r1.y; ar[m2][6] = r1.z; ar[m2][7] = r1.w;
    ai[m2][0] = i0.x; ai[m2][1] = i0.y; ai[m2][2] = i0.z; ai[m2][3] = i0.w;
    ai[m2][4] = i1.x; ai[m2][5] = i1.y; ai[m2][6] = i1.z; ai[m2][7] = i1.w;
  }

  float* bR = sBre[wave];
  float* bI = sBim[wave];
#pragma unroll
  for (int q = 0; q < 8; ++q) {
    const int p = lane * 8 + q;
    const int m1 = p >> 4;
    const float t0r = ar[0][q] + ar[2][q], t0i = ai[0][q] + ai[2][q];
    const float t1r = ar[0][q] - ar[2][q], t1i = ai[0][q] - ai[2][q];
    const float t2r = ar[1][q] + ar[3][q], t2i = ai[1][q] + ai[3][q];
    const float t3r = ar[1][q] - ar[3][q], t3i = ai[1][q] - ai[3][q];
    // y[j2] = 4-point DFT, then *= W64^(m1*j2) = W1024^(16*m1*j2)
    bR[p] = t0r + t2r;
    bI[p] = t0i + t2i;
    {
      const float vr = t1r + t3i, vi = t1i - t3r;  // j2 = 1
      const int t = 16 * m1;
      bR[256 + p] = vr * twc[t] - vi * tws[t];
      bI[256 + p] = vr * tws[t] + vi * twc[t];
    }
    {
      const float vr = t0r - t2r, vi = t0i - t2i;  // j2 = 2
      const int t = 32 * m1;
      bR[512 + p] = vr * twc[t] - vi * tws[t];
      bI[512 + p] = vr * tws[t] + vi * twc[t];
    }
    {
      const float vr = t1r - t3i, vi = t1i + t3r;  // j2 = 3
      const int t = 48 * m1;
      bR[768 + p] = vr * twc[t] - vi * tws[t];
      bI[768 + p] = vr * tws[t] + vi * twc[t];
    }
  }
  __syncthreads();

  // ------------- Stage B: G[.][k2] = F16 x H over m1, + W1024 twiddle -------------
  float* cR = sCre[wave];
  float* cI = sCim[wave];
#pragma unroll
  for (int j2 = 0; j2 < 4; ++j2) {
    v2f hre[4], him[4];
#pragma unroll
    for (int kc = 0; kc < 4; ++kc) {
      const int r0 = 4 * kc + 2 * hi;  // B rows in this lane-half
      const int base = j2 * 256 + m;
      hre[kc].x = bR[base + r0 * 16];
      hre[kc].y = bR[base + (r0 + 1) * 16];
      him[kc].x = bI[base + r0 * 16];
      him[kc].y = bI[base + (r0 + 1) * 16];
    }
    v8f gre = {};
    v8f gim = {};
#pragma unroll
    for (int kc = 0; kc < 4; ++kc) {
      v2f a;  a.x = fre[kc][0];  a.y = fre[kc][1];
      gre = wmma_k4(a, hre[kc], gre);   // +Fre*Hre
      gim = wmma_k4(a, him[kc], gim);   // +Fre*Him
    }
#pragma unroll
    for (int kc = 0; kc < 4; ++kc) {
      v2f an; an.x = fng[kc][0]; an.y = fng[kc][1];
      v2f ap; ap.x = fim[kc][0]; ap.y = fim[kc][1];
      gre = wmma_k4(an, him[kc], gre);  // -Fim*Him
      gim = wmma_k4(ap, hre[kc], gim);  // +Fim*Hre
    }
    // twiddle W1024^(n1*k2) and transpose into [n1][k2] (stride 65)
#pragma unroll
    for (int v = 0; v < 8; ++v) {
      const int j1 = v + 8 * hi;       // D row
      const int k2 = j1 * 4 + j2;
      const int t = m * k2;            // n1 = m
      const float gr = gre[v], gi = gim[v];
      cR[m * 65 + k2] = gr * twc[t] - gi * tws[t];
      cI[m * 65 + k2] = gr * tws[t] + gi * twc[t];
    }
  }
  __syncthreads();

  // ------------- Stage C: X[k1][k2] = F16 x H2 over n1, store -------------
  float* oR = yre + (size_t)row * 1024;
  float* oI = yim + (size_t)row * 1024;
#pragma unroll
  for (int ct = 0; ct < 4; ++ct) {
    const int col = ct * 16 + m;  // k2
    v2f hre[4], him[4];
#pragma unroll
    for (int kc = 0; kc < 4; ++kc) {
      const int r0 = 4 * kc + 2 * hi;  // n1 rows in this lane-half
      hre[kc].x = cR[r0 * 65 + col];
      hre[kc].y = cR[(r0 + 1) * 65 + col];
      him[kc].x = cI[r0 * 65 + col];
      him[kc].y = cI[(r0 + 1) * 65 + col];
    }
    v8f xr = {};
    v8f xi = {};
#pragma unroll
    for (int kc = 0; kc < 4; ++kc) {
      v2f a;  a.x = fre[kc][0];  a.y = fre[kc][1];
      xr = wmma_k4(a, hre[kc], xr);
      xi = wmma_k4(a, him[kc], xi);
    }
#pragma unroll
    for (int kc = 0; kc < 4; ++kc) {
      v2f an; an.x = fng[kc][0]; an.y = fng[kc][1];
      v2f ap; ap.x = fim[kc][0]; ap.y = fim[kc][1];
      xr = wmma_k4(an, him[kc], xr);
      xi = wmma_k4(ap, hre[kc], xi);
    }
#pragma unroll
    for (int v = 0; v < 8; ++v) {
      const int k1 = v + 8 * hi;
      oR[k1 * 64 + col] = xr[v];
      oI[k1 * 64 + col] = xi[v];
    }
  }
}

extern "C" void kernel_launch(void* const* d_in, const int* in_sizes, int n_in,
                              void* d_out, int out_size, void* d_ws,
                              size_t ws_size, hipStream_t stream) {
  const float* xre = (const float*)d_in[0];
  const float* xim = (const float*)d_in[1];
  const int rows = in_sizes[0] / 1024;  // 32768 x 1024 fp32
  float* yre = (float*)d_out;
  float* yim = (float*)d_out + (size_t)rows * 1024;
  const int blocks = (rows + WAVES_PER_BLOCK - 1) / WAVES_PER_BLOCK;
  fft1024_wmma<<<blocks, THREADS, 0, stream>>>(xre, xim, yre, yim, rows);
}